// MuCoMiD_experiment_48567490183490
// MI455X (gfx1250) — compile-verified
//
#include <hip/hip_runtime.h>
#include <hip/hip_bf16.h>

typedef __attribute__((ext_vector_type(2))) float v2f;
typedef __attribute__((ext_vector_type(8))) float v8f;

#define EMBD 128

// ---------------- utility kernels ----------------

__global__ void zero_f32(float* p, long n) {
    long i = (long)blockIdx.x * blockDim.x + threadIdx.x;
    if (i < n) p[i] = 0.0f;
}

// deg[dst] += w  (self-loop +1 added in finalize)
__global__ void edge_deg(const int* __restrict__ el, const float* __restrict__ w,
                         float* __restrict__ deg, int E) {
    int e = blockIdx.x * blockDim.x + threadIdx.x;
    if (e < E) atomicAdd(&deg[el[2 * e + 1]], w[e]);
}

// dinv = rsqrt(deg + 1)
__global__ void fin_dinv(float* __restrict__ deg, int N) {
    int i = blockIdx.x * blockDim.x + threadIdx.x;
    if (i < N) deg[i] = rsqrtf(deg[i] + 1.0f);
}

// out[i,f] = dinv[i]^2 * xw[i,f] + b[f]   (also serves as accumulator init)
__global__ void self_init(const float* __restrict__ xw, const float* __restrict__ dinv,
                          const float* __restrict__ b, float* __restrict__ out, long total) {
    long i = (long)blockIdx.x * blockDim.x + threadIdx.x;
    if (i < total) {
        long n = i >> 7;
        int  f = (int)(i & 127);
        float di = dinv[n];
        out[i] = di * di * xw[i] + b[f];
    }
}

// out[dst,f] += dinv[src]*w*dinv[dst] * xw[src,f]   (2 edges per 256-thread block)
__global__ __launch_bounds__(256)
void edge_scatter(const int* __restrict__ el, const float* __restrict__ w,
                  const float* __restrict__ dinv, const float* __restrict__ xw,
                  float* __restrict__ out, int E) {
    int e = blockIdx.x * 2 + (threadIdx.x >> 7);
    int f = threadIdx.x & 127;
    if (e < E) {
        int s = el[2 * e + 0];
        int d = el[2 * e + 1];
        float norm = dinv[s] * w[e] * dinv[d];
        atomicAdd(&out[(long)d * EMBD + f], norm * xw[(long)s * EMBD + f]);
    }
}

__global__ void relu_f32(float* p, long n) {
    long i = (long)blockIdx.x * blockDim.x + threadIdx.x;
    if (i < n) p[i] = fmaxf(p[i], 0.0f);
}

// ---------------- WMMA f32 GEMM: C[N x 128] = A[N x 128] @ W[128 x 128] ----------------
// Block = 256 threads = 8 waves; wave w owns columns [16w, 16w+16) of a 16-row tile.
// V_WMMA_F32_16X16X4_F32 per ISA 7.12.2:
//   A 16x4: lanes 0-15 -> M, vgpr0 = {K0 | K2 across lane halves}, vgpr1 = {K1 | K3}
//   B 4x16 mirrored; C/D 16x16: vgpr v -> M = v + 8*half, N = lane%16.
__global__ __launch_bounds__(256)
void gemm_n128_wmma(const float* __restrict__ A, const float* __restrict__ W,
                    float* __restrict__ C, int N) {
    const int wave = threadIdx.x >> 5;
    const int lane = threadIdx.x & 31;
    const int half = lane >> 4;
    const int l    = lane & 15;
    const long row0 = (long)blockIdx.x * 16;
    const int  col0 = wave * 16;

    long arow = row0 + l;
    if (arow >= N) arow = N - 1;          // clamp reads; EXEC stays all-1s for WMMA
    const float* __restrict__ ap = A + arow * EMBD;

    v8f acc = {};
#pragma unroll
    for (int kb = 0; kb < EMBD; kb += 4) {
        const int k = kb + 2 * half;
        v2f a, b;
        a.x = ap[k];
        a.y = ap[k + 1];
        b.x = W[(long)k * EMBD + col0 + l];
        b.y = W[(long)(k + 1) * EMBD + col0 + l];
        acc = __builtin_amdgcn_wmma_f32_16x16x4_f32(
            /*neg_a=*/false, a, /*neg_b=*/false, b,
            /*c_mod=*/(short)0, acc, /*reuse_a=*/false, /*reuse_b=*/false);
    }
#pragma unroll
    for (int v = 0; v < 8; ++v) {
        long r = row0 + v + 8 * half;
        if (r < N) C[r * EMBD + col0 + l] = acc[v];
    }
}

// ---------------- pair heads: sigmoid(dot(ha[i]*hb[j], W) + b) ----------------
// One wave32 per pair; each lane handles exactly one float4 (32*4 = 128 features).
__global__ __launch_bounds__(256)
void pair_head(const int* __restrict__ pairs, const float* __restrict__ ha,
               const float* __restrict__ hb, const float* __restrict__ Wv,
               const float* __restrict__ bptr, float* __restrict__ out, int P) {
    int wid  = (int)(((long)blockIdx.x * blockDim.x + threadIdx.x) >> 5);
    int lane = threadIdx.x & 31;
    if (wid >= P) return;
    int i = pairs[2 * wid + 0];
    int j = pairs[2 * wid + 1];
    const float4* a = (const float4*)(ha + (long)i * EMBD);
    const float4* b = (const float4*)(hb + (long)j * EMBD);
    const float4* w = (const float4*)Wv;
    float4 av = a[lane], bv = b[lane], wv = w[lane];
    float s = av.x * bv.x * wv.x + av.y * bv.y * wv.y +
              av.z * bv.z * wv.z + av.w * bv.w * wv.w;
#pragma unroll
    for (int off = 16; off > 0; off >>= 1) s += __shfl_xor(s, off, 32);
    if (lane == 0) out[wid] = 1.0f / (1.0f + __expf(-(s + bptr[0])));
}

// ---------------- host-side orchestration ----------------

static inline int cdiv(long a, long b) { return (int)((a + b - 1) / b); }

static void run_gcn_graph(const float* emb, int N, const int* el, const float* w, int E,
                          const float* W0, const float* b0, const float* W1, const float* b1,
                          float* hid, float* dinv, float* xw, float* h1, hipStream_t stream) {
    const long feats = (long)N * EMBD;
    // degree -> dinv (graph-invariant across layers)
    zero_f32<<<cdiv(N, 256), 256, 0, stream>>>(dinv, N);
    edge_deg<<<cdiv(E, 256), 256, 0, stream>>>(el, w, dinv, E);
    fin_dinv<<<cdiv(N, 256), 256, 0, stream>>>(dinv, N);
    // layer 0: emb @ W0 -> xw ; aggregate -> h1
    gemm_n128_wmma<<<cdiv(N, 16), 256, 0, stream>>>(emb, W0, xw, N);
    self_init<<<cdiv(feats, 256), 256, 0, stream>>>(xw, dinv, b0, h1, feats);
    edge_scatter<<<cdiv(E, 2), 256, 0, stream>>>(el, w, dinv, xw, h1, E);
    // layer 1: h1 @ W1 -> xw ; aggregate -> hid ; relu
    gemm_n128_wmma<<<cdiv(N, 16), 256, 0, stream>>>(h1, W1, xw, N);
    self_init<<<cdiv(feats, 256), 256, 0, stream>>>(xw, dinv, b1, hid, feats);
    edge_scatter<<<cdiv(E, 2), 256, 0, stream>>>(el, w, dinv, xw, hid, E);
    relu_f32<<<cdiv(feats, 256), 256, 0, stream>>>(hid, feats);
}

extern "C" void kernel_launch(void* const* d_in, const int* in_sizes, int n_in,
                              void* d_out, int out_size, void* d_ws, size_t ws_size,
                              hipStream_t stream) {
    const float* memb = (const float*)d_in[0];
    const float* demb = (const float*)d_in[1];
    const float* pemb = (const float*)d_in[2];
    const int*   mel  = (const int*)d_in[3];
    const float* mw   = (const float*)d_in[4];
    const int*   del  = (const int*)d_in[5];
    const float* dw   = (const float*)d_in[6];
    const int*   pel  = (const int*)d_in[7];
    const float* pw   = (const float*)d_in[8];
    const int* mp_pairs = (const int*)d_in[9];
    const int* dp_pairs = (const int*)d_in[10];
    const int* md_pairs = (const int*)d_in[11];
    const float *mW0 = (const float*)d_in[12], *mb0 = (const float*)d_in[13];
    const float *mW1 = (const float*)d_in[14], *mb1 = (const float*)d_in[15];
    const float *dW0 = (const float*)d_in[16], *db0 = (const float*)d_in[17];
    const float *dW1 = (const float*)d_in[18], *db1 = (const float*)d_in[19];
    const float *pW0 = (const float*)d_in[20], *pb0 = (const float*)d_in[21];
    const float *pW1 = (const float*)d_in[22], *pb1 = (const float*)d_in[23];
    const float *assoc_W = (const float*)d_in[24], *assoc_b = (const float*)d_in[25];
    const float *mp_W    = (const float*)d_in[26], *mp_b    = (const float*)d_in[27];
    const float *dp_W    = (const float*)d_in[28], *dp_b    = (const float*)d_in[29];

    const int Nm = in_sizes[0] / EMBD, Nd = in_sizes[1] / EMBD, Np = in_sizes[2] / EMBD;
    const int Em = in_sizes[3] / 2,    Ed = in_sizes[5] / 2,    Ep = in_sizes[7] / 2;
    const int Pmp = in_sizes[9] / 2,   Pdp = in_sizes[10] / 2,  Pmd = in_sizes[11] / 2;
    int Nmax = Nm > Nd ? Nm : Nd; if (Np > Nmax) Nmax = Np;

    // workspace layout (floats): persistent hiddens then shared transients
    float* ws    = (float*)d_ws;
    float* hid_m = ws;
    float* hid_d = hid_m + (size_t)Nm * EMBD;
    float* hid_p = hid_d + (size_t)Nd * EMBD;
    float* dinv  = hid_p + (size_t)Np * EMBD;
    float* xw    = dinv  + (size_t)Nmax;
    float* h1    = xw    + (size_t)Nmax * EMBD;

    run_gcn_graph(memb, Nm, mel, mw, Em, mW0, mb0, mW1, mb1, hid_m, dinv, xw, h1, stream);
    run_gcn_graph(demb, Nd, del, dw, Ed, dW0, db0, dW1, db1, hid_d, dinv, xw, h1, stream);
    run_gcn_graph(pemb, Np, pel, pw, Ep, pW0, pb0, pW1, pb1, hid_p, dinv, xw, h1, stream);

    // outputs: assoc (md), mp, dp — concatenated in reference return order
    float* out = (float*)d_out;
    pair_head<<<cdiv((long)Pmd * 32, 256), 256, 0, stream>>>(md_pairs, hid_m, hid_d, assoc_W, assoc_b, out, Pmd);
    pair_head<<<cdiv((long)Pmp * 32, 256), 256, 0, stream>>>(mp_pairs, hid_m, hid_p, mp_W, mp_b, out + Pmd, Pmp);
    pair_head<<<cdiv((long)Pdp * 32, 256), 256, 0, stream>>>(dp_pairs, hid_d, hid_p, dp_W, dp_b, out + Pmd + Pmp, Pdp);
}